// GraphSAGE_14053132992904
// MI455X (gfx1250) — compile-verified
//
#include <hip/hip_runtime.h>

typedef __attribute__((ext_vector_type(2))) float v2f;
typedef __attribute__((ext_vector_type(8))) float v8f;

#define N_NODES 100000
#define N_EDGES 500000
#define DIM 128

// ---------------------------------------------------------------------------
// Zero a float buffer (grid-stride).
// ---------------------------------------------------------------------------
__global__ void zero_f32(float* __restrict__ p, long n) {
  long i = (long)blockIdx.x * blockDim.x + threadIdx.x;
  long stride = (long)gridDim.x * blockDim.x;
  for (; i < n; i += stride) p[i] = 0.0f;
}

// ---------------------------------------------------------------------------
// Edge scatter: one wave32 per edge. Each lane loads float4 (4 dims) of the
// source-node feature row and atomically adds into the destination row.
// agg[dst] += feat[src]; optionally cnt[dst] += 1.
// L2 is 192MB; the 51.2MB accumulator stays resident, so the 64M f32 atomics
// per layer are L2-local.
// ---------------------------------------------------------------------------
template <bool COUNT>
__global__ void scatter_mean_accum(const float* __restrict__ feat,
                                   const int* __restrict__ src,
                                   const int* __restrict__ dst,
                                   float* __restrict__ agg,
                                   float* __restrict__ cnt) {
  int lane = threadIdx.x & 31;
  int edge = blockIdx.x * (blockDim.x >> 5) + (threadIdx.x >> 5);
  if (edge >= N_EDGES) return;
  int s = src[edge];
  int d = dst[edge];
  const float4* fs = (const float4*)(feat + (long)s * DIM);
  float4 v = fs[lane];
  float* ad = agg + (long)d * DIM + lane * 4;
  atomicAdd(ad + 0, v.x);
  atomicAdd(ad + 1, v.y);
  atomicAdd(ad + 2, v.z);
  atomicAdd(ad + 3, v.w);
  if (COUNT && lane == 0) atomicAdd(cnt + d, 1.0f);
}

// ---------------------------------------------------------------------------
// Fused SAGE layer GEMM:
//   out = act( (agg / max(cnt,1)) @ Wl + bias + self_feat @ Wr )
// One block = 16 rows x 128 cols of output. 8 waves; wave w owns the 16x16
// tile at columns [16w, 16w+16). Per wave: 2 x (K=128 / 4) = 64
// v_wmma_f32_16x16x4_f32 ops accumulating in 8 f32 VGPRs.
//
// VGPR layouts per CDNA5 ISA 7.12.2 (16x16x4 f32):
//   A (16x4): lanes 0-15 hold row M=lane, K={k, k+1}; lanes 16-31 hold
//             row M=lane-16, K={k+2, k+3}  -> contiguous float2 per lane.
//   B (4x16): lanes 0-15 hold col N=lane, K rows {k, k+1}; lanes 16-31 hold
//             col N=lane-16, K rows {k+2, k+3} -> two strided b32 loads.
//   C/D (16x16): VGPR r, lanes 0-15 = (M=r, N=lane); lanes 16-31 =
//             (M=r+8, N=lane-16).
// M = 100000 is a multiple of 16 -> no tail, EXEC all-ones (WMMA requirement).
// ---------------------------------------------------------------------------
template <bool RELU>
__global__ void sage_gemm(const float* __restrict__ agg,
                          const float* __restrict__ cnt,
                          const float* __restrict__ self_feat,
                          const float* __restrict__ Wl,
                          const float* __restrict__ Wr,
                          const float* __restrict__ bias,
                          float* __restrict__ out) {
  const int lane  = threadIdx.x & 31;
  const int wave  = threadIdx.x >> 5;   // 0..7 -> N tile
  const int m16   = lane & 15;
  const int khalf = lane >> 4;          // 0 or 1
  const int koff  = khalf * 2;
  const int row   = blockIdx.x * 16 + m16;   // A row this lane feeds
  const int col   = wave * 16 + m16;         // B/C/D column this lane feeds

  const float inv = 1.0f / fmaxf(cnt[row], 1.0f);

  // Init accumulator with the bias (added exactly once via the C operand).
  const float bv = bias[col];
  v8f acc;
#pragma unroll
  for (int r = 0; r < 8; ++r) acc[r] = bv;

  const float* arow = agg + (long)row * DIM;
  const float* xrow = self_feat + (long)row * DIM;

  // Pass 1: (agg * inv) @ Wl   (mean-divide fused into the A fragment)
#pragma unroll 4
  for (int k = 0; k < DIM; k += 4) {
    v2f a = *(const v2f*)(arow + k + koff);
    a.x *= inv;
    a.y *= inv;
    v2f b;
    b.x = Wl[(k + koff) * DIM + col];
    b.y = Wl[(k + koff + 1) * DIM + col];
    acc = __builtin_amdgcn_wmma_f32_16x16x4_f32(
        false, a, false, b, (short)0, acc, false, false);
  }

  // Pass 2: self_feat @ Wr
#pragma unroll 4
  for (int k = 0; k < DIM; k += 4) {
    v2f a = *(const v2f*)(xrow + k + koff);
    v2f b;
    b.x = Wr[(k + koff) * DIM + col];
    b.y = Wr[(k + koff + 1) * DIM + col];
    acc = __builtin_amdgcn_wmma_f32_16x16x4_f32(
        false, a, false, b, (short)0, acc, false, false);
  }

  // Store D: VGPR r -> row (tileM + khalf*8 + r), column col.
  long obase = ((long)blockIdx.x * 16 + khalf * 8) * DIM + col;
#pragma unroll
  for (int r = 0; r < 8; ++r) {
    float v = acc[r];
    if (RELU) v = fmaxf(v, 0.0f);
    out[obase + (long)r * DIM] = v;
  }
}

// ---------------------------------------------------------------------------
// Host-side launcher. Workspace layout (bytes):
//   [0,            51,200,000)  agg   (N_NODES*DIM f32, reused by both layers)
//   [51,200,000,   51,600,000)  cnt   (N_NODES f32)
//   [51,609,600,  102,809,600)  h     (N_NODES*DIM f32, layer-1 output)
// ---------------------------------------------------------------------------
extern "C" void kernel_launch(void* const* d_in, const int* in_sizes, int n_in,
                              void* d_out, int out_size, void* d_ws, size_t ws_size,
                              hipStream_t stream) {
  const float* x   = (const float*)d_in[0];
  const int*   ei  = (const int*)d_in[1];
  const float* Wl1 = (const float*)d_in[2];
  const float* bl1 = (const float*)d_in[3];
  const float* Wr1 = (const float*)d_in[4];
  const float* Wl2 = (const float*)d_in[5];
  const float* bl2 = (const float*)d_in[6];
  const float* Wr2 = (const float*)d_in[7];
  float* out = (float*)d_out;

  const int* src = ei;             // edge_index[0]
  const int* dst = ei + N_EDGES;   // edge_index[1]

  char* ws = (char*)d_ws;
  const size_t aggBytes = (size_t)N_NODES * DIM * sizeof(float);  // 51,200,000
  float* agg = (float*)ws;
  float* cnt = (float*)(ws + aggBytes);
  float* h   = (float*)(ws + aggBytes + 409600);  // 256B-aligned past cnt

  const int edgeBlocks = N_EDGES / 8;   // 8 edges per 256-thread block (exact)
  const int gemmBlocks = N_NODES / 16;  // 6250 (exact)

  // ---- Layer 1 ----
  zero_f32<<<2048, 256, 0, stream>>>(agg, (long)N_NODES * DIM + N_NODES);
  scatter_mean_accum<true><<<edgeBlocks, 256, 0, stream>>>(x, src, dst, agg, cnt);
  sage_gemm<true><<<gemmBlocks, 256, 0, stream>>>(agg, cnt, x, Wl1, Wr1, bl1, h);

  // ---- Layer 2 ----
  zero_f32<<<2048, 256, 0, stream>>>(agg, (long)N_NODES * DIM);
  scatter_mean_accum<false><<<edgeBlocks, 256, 0, stream>>>(h, src, dst, agg, cnt);
  sage_gemm<false><<<gemmBlocks, 256, 0, stream>>>(agg, cnt, h, Wl2, Wr2, bl2, out);
}